// PointTransformerNet_47115791237239
// MI455X (gfx1250) — compile-verified
//
#include <hip/hip_runtime.h>
#include <math.h>

// ---------------------------------------------------------------------------
// PointTransformerConv x2 + MLP head for gfx1250 (MI455X).
// f32 WMMA (V_WMMA_F32_16X16X4_F32) for all node-level GEMMs; edge softmax
// via float->uint monotone-key atomicMax + fused num/den atomicAdd pass.
// Round 3: float2 (b64) edge gathers, float4 (b128) finalize.
// ---------------------------------------------------------------------------

typedef float v2f __attribute__((ext_vector_type(2)));
typedef float v8f __attribute__((ext_vector_type(8)));

__device__ __forceinline__ v8f wmma4(v2f a, v2f b, v8f c) {
  // D(16x16,f32) = A(16x4,f32) * B(4x16,f32) + C
  return __builtin_amdgcn_wmma_f32_16x16x4_f32(
      /*neg_a=*/false, a, /*neg_b=*/false, b,
      /*c_mod=*/(short)0, c, /*reuse_a=*/false, /*reuse_b=*/false);
}

// monotone float -> uint key (order preserved) for atomic max on f32
__device__ __forceinline__ unsigned fkey(float f) {
  unsigned b = __float_as_uint(f);
  return (b & 0x80000000u) ? ~b : (b | 0x80000000u);
}
__device__ __forceinline__ float fdec(unsigned k) {
  return __uint_as_float((k & 0x80000000u) ? (k ^ 0x80000000u) : ~k);
}

// ---- conv1 projections: X[N,3] @ {Ws,Wd,Wv}[3,64]  (K padded 3->4) --------
__global__ __launch_bounds__(256) void k_proj3(
    const float* __restrict__ X,
    const float* __restrict__ Ws, const float* __restrict__ Wd,
    const float* __restrict__ Wv,
    float* __restrict__ As, float* __restrict__ Ad, float* __restrict__ Vv,
    int n) {
  const int lane = threadIdx.x & 31;
  const int wave = threadIdx.x >> 5;
  const int hi = lane >> 4;          // lanes 16..31 hold K=2,3
  const int lm = lane & 15;
  const int base = (blockIdx.x * 8 + wave) * 16;
  if (base >= n) return;             // wave-uniform: EXEC stays all-ones
  const bool full = (base + 16 <= n);  // wave-uniform (always true if n%16==0)

  const int row = base + lm;
  const int rr = (row < n) ? row : (n - 1);
  // unconditional loads + select (no predicated-load exec juggling)
  const float x0 = X[rr * 3 + 2 * hi];   // K = 0 or 2 (both < 3)
  const float x1 = X[rr * 3 + 1];
  v2f a;
  a.x = x0;
  a.y = hi ? 0.0f : x1;                  // K = 1 valid, K = 3 padded 0

  const float* Wm[3] = {Ws, Wd, Wv};
  float* Om[3] = {As, Ad, Vv};
#pragma unroll
  for (int m = 0; m < 3; ++m) {
#pragma unroll
    for (int ct = 0; ct < 4; ++ct) {
      const int col = ct * 16 + lm;
      const float w0 = Wm[m][(2 * hi) * 64 + col];
      const float w1 = Wm[m][64 + col];
      v2f b;
      b.x = w0;
      b.y = hi ? 0.0f : w1;
      v8f c = {};
      c = wmma4(a, b, c);
      float* o = Om[m] + (size_t)(base + 8 * hi) * 64 + col;
      if (full) {
#pragma unroll
        for (int r = 0; r < 8; ++r) o[r * 64] = c[r];
      } else {
#pragma unroll
        for (int r = 0; r < 8; ++r)
          if (base + 8 * hi + r < n) o[r * 64] = c[r];
      }
    }
  }
}

// ---- conv2 projections: X[N,64] @ {Ws,Wd,Wv}[64,64], weights in LDS -------
__global__ __launch_bounds__(256) void k_proj64(
    const float* __restrict__ X,
    const float* __restrict__ Ws, const float* __restrict__ Wd,
    const float* __restrict__ Wv,
    float* __restrict__ As, float* __restrict__ Ad, float* __restrict__ Vv,
    int n) {
  __shared__ float w[3][64 * 64];   // 48 KB of the 320 KB WGP LDS
  for (int i = threadIdx.x; i < 4096; i += 256) {
    w[0][i] = Ws[i]; w[1][i] = Wd[i]; w[2][i] = Wv[i];
  }
  __syncthreads();

  const int lane = threadIdx.x & 31;
  const int wave = threadIdx.x >> 5;
  const int hi = lane >> 4;
  const int lm = lane & 15;
  const int base = (blockIdx.x * 8 + wave) * 16;
  if (base >= n) return;
  const bool full = (base + 16 <= n);

  const int row = base + lm;
  const int rr = (row < n) ? row : (n - 1);
  const float* xr = X + (size_t)rr * 64;

  v8f acc[3][4] = {};
  for (int k0 = 0; k0 < 64; k0 += 4) {
    const int kb = k0 + 2 * hi;
    const v2f a = *(const v2f*)(xr + kb);   // 8B-aligned -> global_load_b64
#pragma unroll
    for (int m = 0; m < 3; ++m) {
#pragma unroll
      for (int ct = 0; ct < 4; ++ct) {
        const int col = ct * 16 + lm;
        v2f b;
        b.x = w[m][kb * 64 + col];          // bank = col: conflict-free
        b.y = w[m][(kb + 1) * 64 + col];
        acc[m][ct] = wmma4(a, b, acc[m][ct]);
      }
    }
  }

  float* Om[3] = {As, Ad, Vv};
#pragma unroll
  for (int m = 0; m < 3; ++m)
#pragma unroll
    for (int ct = 0; ct < 4; ++ct) {
      float* o = Om[m] + (size_t)(base + 8 * hi) * 64 + ct * 16 + lm;
      if (full) {
#pragma unroll
        for (int r = 0; r < 8; ++r) o[r * 64] = acc[m][ct][r];
      } else {
#pragma unroll
        for (int r = 0; r < 8; ++r)
          if (base + 8 * hi + r < n) o[r * 64] = acc[m][ct][r];
      }
    }
}

// ---- edge pass 1: alpha + segment max (atomic on monotone uint key) -------
// lane -> channel pair (2*lane, 2*lane+1): all row gathers are 8B float2.
__global__ __launch_bounds__(256) void k_edge_max(
    const int* __restrict__ ei, int E_,
    const float* __restrict__ pos,
    const float* __restrict__ Asrc, const float* __restrict__ Adst,
    const float* __restrict__ Wp, const float* __restrict__ bp,
    unsigned* __restrict__ mkey) {
  const int gw = (int)((blockIdx.x * 256u + threadIdx.x) >> 5);  // wave/edge
  const int c2 = (threadIdx.x & 31) * 2;
  if (gw >= E_) return;
  const int sj = ei[gw];
  const int di = ei[E_ + gw];
  const float r0 = pos[di * 3 + 0] - pos[sj * 3 + 0];
  const float r1 = pos[di * 3 + 1] - pos[sj * 3 + 1];
  const float r2 = pos[di * 3 + 2] - pos[sj * 3 + 2];

  const float2 wp0 = *(const float2*)(Wp + c2);
  const float2 wp1 = *(const float2*)(Wp + 64 + c2);
  const float2 wp2 = *(const float2*)(Wp + 128 + c2);
  const float2 bpv = *(const float2*)(bp + c2);
  const size_t db = (size_t)di * 64 + c2;
  const float2 ad = *(const float2*)(Adst + db);
  const float2 as = *(const float2*)(Asrc + (size_t)sj * 64 + c2);

  const float d0 = bpv.x + r0 * wp0.x + r1 * wp1.x + r2 * wp2.x;
  const float d1 = bpv.y + r0 * wp0.y + r1 * wp1.y + r2 * wp2.y;
  atomicMax(&mkey[db],     fkey(ad.x - as.x + d0));
  atomicMax(&mkey[db + 1], fkey(ad.y - as.y + d1));
}

// ---- edge pass 2: e = exp(alpha-m); fused num/den scatter -----------------
__global__ __launch_bounds__(256) void k_edge_acc(
    const int* __restrict__ ei, int E_,
    const float* __restrict__ pos,
    const float* __restrict__ Asrc, const float* __restrict__ Adst,
    const float* __restrict__ Vv,
    const float* __restrict__ Wp, const float* __restrict__ bp,
    const unsigned* __restrict__ mkey,
    float* __restrict__ Ssum, float* __restrict__ Num) {
  const int gw = (int)((blockIdx.x * 256u + threadIdx.x) >> 5);
  const int c2 = (threadIdx.x & 31) * 2;
  if (gw >= E_) return;
  const int sj = ei[gw];
  const int di = ei[E_ + gw];
  const float r0 = pos[di * 3 + 0] - pos[sj * 3 + 0];
  const float r1 = pos[di * 3 + 1] - pos[sj * 3 + 1];
  const float r2 = pos[di * 3 + 2] - pos[sj * 3 + 2];

  const float2 wp0 = *(const float2*)(Wp + c2);
  const float2 wp1 = *(const float2*)(Wp + 64 + c2);
  const float2 wp2 = *(const float2*)(Wp + 128 + c2);
  const float2 bpv = *(const float2*)(bp + c2);
  const size_t sb = (size_t)sj * 64 + c2;
  const size_t db = (size_t)di * 64 + c2;
  const float2 ad = *(const float2*)(Adst + db);
  const float2 as = *(const float2*)(Asrc + sb);
  const float2 vv = *(const float2*)(Vv + sb);
  const uint2  mk = *(const uint2*)(mkey + db);

  const float d0 = bpv.x + r0 * wp0.x + r1 * wp1.x + r2 * wp2.x;
  const float d1 = bpv.y + r0 * wp0.y + r1 * wp1.y + r2 * wp2.y;
  const float e0 = expf((ad.x - as.x + d0) - fdec(mk.x));
  const float e1 = expf((ad.y - as.y + d1) - fdec(mk.y));
  atomicAdd(&Ssum[db],     e0);
  atomicAdd(&Ssum[db + 1], e1);
  atomicAdd(&Num[db],      e0 * (vv.x + d0));
  atomicAdd(&Num[db + 1],  e1 * (vv.y + d1));
}

// ---- node finalize: relu(num / (den + eps)), b128 vectorized --------------
__global__ __launch_bounds__(256) void k_finalize(
    const float* __restrict__ Num, const float* __restrict__ Ssum,
    float* __restrict__ H, int total4) {
  const int i = blockIdx.x * 256 + threadIdx.x;
  if (i >= total4) return;
  const float4 nu = ((const float4*)Num)[i];
  const float4 de = ((const float4*)Ssum)[i];
  float4 o;
  o.x = fmaxf(nu.x / (de.x + 1e-16f), 0.0f);
  o.y = fmaxf(nu.y / (de.y + 1e-16f), 0.0f);
  o.z = fmaxf(nu.z / (de.z + 1e-16f), 0.0f);
  o.w = fmaxf(nu.w / (de.w + 1e-16f), 0.0f);
  ((float4*)H)[i] = o;
}

// ---- head fc1: H2[N,64] @ W[64,32] + b, relu ------------------------------
__global__ __launch_bounds__(256) void k_fc1(
    const float* __restrict__ X, const float* __restrict__ W,
    const float* __restrict__ bias, float* __restrict__ Y, int n) {
  __shared__ float w[64 * 32];
  for (int i = threadIdx.x; i < 2048; i += 256) w[i] = W[i];
  __syncthreads();

  const int lane = threadIdx.x & 31;
  const int wave = threadIdx.x >> 5;
  const int hi = lane >> 4;
  const int lm = lane & 15;
  const int base = (blockIdx.x * 8 + wave) * 16;
  if (base >= n) return;
  const bool full = (base + 16 <= n);

  const int row = base + lm;
  const int rr = (row < n) ? row : (n - 1);
  const float* xr = X + (size_t)rr * 64;

  v8f acc[2] = {};
  for (int k0 = 0; k0 < 64; k0 += 4) {
    const int kb = k0 + 2 * hi;
    const v2f a = *(const v2f*)(xr + kb);
#pragma unroll
    for (int ct = 0; ct < 2; ++ct) {
      const int col = ct * 16 + lm;
      v2f b;
      b.x = w[kb * 32 + col];
      b.y = w[(kb + 1) * 32 + col];
      acc[ct] = wmma4(a, b, acc[ct]);
    }
  }
#pragma unroll
  for (int ct = 0; ct < 2; ++ct) {
    const int col = ct * 16 + lm;
    const float bb = bias[col];
    float* o = Y + (size_t)(base + 8 * hi) * 32 + col;
    if (full) {
#pragma unroll
      for (int r = 0; r < 8; ++r) o[r * 32] = fmaxf(acc[ct][r] + bb, 0.0f);
    } else {
#pragma unroll
      for (int r = 0; r < 8; ++r)
        if (base + 8 * hi + r < n) o[r * 32] = fmaxf(acc[ct][r] + bb, 0.0f);
    }
  }
}

// ---- head fc2: [N,32] @ [32,1] + b ----------------------------------------
__global__ __launch_bounds__(256) void k_fc2(
    const float* __restrict__ H3, const float* __restrict__ W,
    const float* __restrict__ b, float* __restrict__ out, int n) {
  const int i = blockIdx.x * 256 + threadIdx.x;
  if (i >= n) return;
  float s = b[0];
  const float* h = H3 + (size_t)i * 32;
#pragma unroll
  for (int j = 0; j < 32; ++j) s += h[j] * W[j];
  out[i] = s;
}

// ---------------------------------------------------------------------------
extern "C" void kernel_launch(void* const* d_in, const int* in_sizes, int n_in,
                              void* d_out, int out_size, void* d_ws,
                              size_t ws_size, hipStream_t stream) {
  const float* x   = (const float*)d_in[0];
  const float* pos = (const float*)d_in[1];
  const int*   ei  = (const int*)d_in[2];
  // d_in[3] = batch (unused, single graph)
  const float* W1s = (const float*)d_in[4];
  const float* W1d = (const float*)d_in[5];
  const float* W1v = (const float*)d_in[6];
  const float* W1p = (const float*)d_in[7];
  const float* b1p = (const float*)d_in[8];
  const float* W2s = (const float*)d_in[9];
  const float* W2d = (const float*)d_in[10];
  const float* W2v = (const float*)d_in[11];
  const float* W2p = (const float*)d_in[12];
  const float* b2p = (const float*)d_in[13];
  const float* Wf1 = (const float*)d_in[14];
  const float* bf1 = (const float*)d_in[15];
  const float* Wf2 = (const float*)d_in[16];
  const float* bf2 = (const float*)d_in[17];

  const int n = in_sizes[0] / 3;   // 50000
  const int E = in_sizes[2] / 2;   // 800000
  const size_t N64 = (size_t)n * 64;

  // workspace layout (floats), 6 * N64 * 4 bytes total (~77 MB):
  // [0]=a_src [1]=a_dst [2]=v [3]=maxkey/H [4]=den/H3 [5]=num
  float*    ws_f = (float*)d_ws;
  float*    As   = ws_f;
  float*    Ad   = ws_f + N64;
  float*    Vv   = ws_f + 2 * N64;
  unsigned* MK   = (unsigned*)(ws_f + 3 * N64);
  float*    Ssum = ws_f + 4 * N64;
  float*    Num  = ws_f + 5 * N64;
  float*    H    = (float*)MK;   // conv output overwrites max-key slot
  float*    H3   = Ssum;         // fc1 output overwrites denominator slot

  const dim3 b256(256);
  const int projBlocks = (n + 127) / 128;
  const int edgeBlocks = (E + 7) / 8;        // one wave32 per edge
  const int total4    = (int)(N64 / 4);
  const int finBlocks  = (total4 + 255) / 256;
  const int fc2Blocks  = (n + 255) / 256;

  // ---- conv1 ----
  hipMemsetAsync(MK, 0, 3 * N64 * sizeof(float), stream);  // MK, den, num
  k_proj3<<<projBlocks, b256, 0, stream>>>(x, W1s, W1d, W1v, As, Ad, Vv, n);
  k_edge_max<<<edgeBlocks, b256, 0, stream>>>(ei, E, pos, As, Ad, W1p, b1p, MK);
  k_edge_acc<<<edgeBlocks, b256, 0, stream>>>(ei, E, pos, As, Ad, Vv, W1p, b1p,
                                              MK, Ssum, Num);
  k_finalize<<<finBlocks, b256, 0, stream>>>(Num, Ssum, H, total4);

  // ---- conv2 ----
  k_proj64<<<projBlocks, b256, 0, stream>>>(H, W2s, W2d, W2v, As, Ad, Vv, n);
  hipMemsetAsync(MK, 0, 3 * N64 * sizeof(float), stream);  // H consumed above
  k_edge_max<<<edgeBlocks, b256, 0, stream>>>(ei, E, pos, As, Ad, W2p, b2p, MK);
  k_edge_acc<<<edgeBlocks, b256, 0, stream>>>(ei, E, pos, As, Ad, Vv, W2p, b2p,
                                              MK, Ssum, Num);
  k_finalize<<<finBlocks, b256, 0, stream>>>(Num, Ssum, H, total4);

  // ---- MLP head ----
  k_fc1<<<projBlocks, b256, 0, stream>>>(H, Wf1, bf1, H3, n);
  k_fc2<<<fc2Blocks, b256, 0, stream>>>(H3, Wf2, bf2, (float*)d_out, n);
}